// HeteroGATEncoder_36679020708478
// MI455X (gfx1250) — compile-verified
//
#include <hip/hip_runtime.h>
#include <hip/hip_bf16.h>
#include <math.h>
#include <stdint.h>

#define NNODE 50000
#define HIDN  128
#define NEDGE 500000

typedef float v2f __attribute__((ext_vector_type(2)));
typedef float v8f __attribute__((ext_vector_type(8)));
typedef unsigned int v4u __attribute__((ext_vector_type(4)));
typedef int v4i __attribute__((ext_vector_type(4)));
typedef int v8i __attribute__((ext_vector_type(8)));

#define HAVE_TDM __has_builtin(__builtin_amdgcn_tensor_load_to_lds)

// monotone float->int encoding so integer atomicMax == float max
__device__ __forceinline__ int enc_f32(float f) {
  int b = __float_as_int(f);
  return b >= 0 ? b : (b ^ 0x7fffffff);
}
__device__ __forceinline__ float dec_f32(int k) {
  return __int_as_float(k >= 0 ? k : (k ^ 0x7fffffff));
}
#define ENC_NEG_INF ((int)0x807fffff)   // enc(-inf)

__device__ __forceinline__ void atom_add_f32(float* p, float v) {
  // lowers to global_atomic_add_f32 (L2 atomic) on gfx1250
  __hip_atomic_fetch_add(p, v, __ATOMIC_RELAXED, __HIP_MEMORY_SCOPE_AGENT);
}

// ---------------------------------------------------------------------------
// Y[N,128] = X[N,K] @ W[K,128] (+bias) using V_WMMA_F32_16X16X4_F32.
// Block = 256 threads = 8 waves; block owns a 16-row tile, wave w owns cols
// [16w,16w+16). A-tile is staged into LDS by the Tensor Data Mover (TDM):
// 2D descriptor, tile 16 x K, with TDM pad (2 DWORDs per K DWORDs) matching
// the lda = K+2 anti-bank-conflict layout the WMMA loop reads.
//
// Dependency modeling: the TDM's LDS writes are invisible to the compiler
// (the LDS address is descriptor *data*).  A bare "memory" clobber was NOT
// sufficient (round 3): sA's address never escaped, so GlobalsAA proved no
// asm/call could write it and folded the never-stored LDS loads to undef
// (constant v[0:1] A operand).  Passing &sA[0] as an asm operand makes the
// global address-taken, so the clobber genuinely aliases sA and the fragment
// loads must be (re)issued after the TDM each iteration.
//
// A fragment (16x4 f32): lane m=lane&15, half=lane>>4, elem j <-> K=2*half+j.
// B fragment symmetric with n=lane&15 (read from global; W is L0/L2-resident).
// C/D: VGPR r <-> row M = r + 8*half, col N = lane&15.
// ---------------------------------------------------------------------------
__global__ __launch_bounds__(256) void gemm_wmma_f32(
    const float* __restrict__ X, const float* __restrict__ W,
    const float* __restrict__ bias, float* __restrict__ Y,
    int N, int K)
{
  __shared__ float sA[16 * 130];            // 16 rows, stride K+2 (pad, K<=128)
  const int lane = threadIdx.x & 31;
  const int wave = threadIdx.x >> 5;
  const int mn   = lane & 15;
  const int half = lane >> 4;
  const int col0 = wave * 16;
  const int lda  = K + 2;
  const int ntile = N >> 4;

  for (int tile = blockIdx.x; tile < ntile; tile += (int)gridDim.x) {
    const int row0 = tile << 4;

#if HAVE_TDM
    // escape sA into the asm + memory clobber: orders prior-iteration reads
    asm volatile("" :: "v"(&sA[0]) : "memory");
    if (wave == 0) {
      // ---- Tensor DMA: global X[row0:row0+16, 0:K] -> LDS sA (padded rows)
      unsigned long long ga =
          (unsigned long long)(uintptr_t)X + (unsigned long long)row0 * K * 4ull;
      v4u g0;
      g0[0] = 1u;                                   // count=1, no gather
      g0[1] = 0u;                                   // lds_addr = offset of sA
      g0[2] = (unsigned)ga;                         // global_addr[31:0]
      g0[3] = (unsigned)(ga >> 32) | (2u << 30);    // addr[56:32] | type=2
      const unsigned pi = (K == 128) ? 6u : 5u;     // pad every 128/64 DWORDs
      v8i g1;
      g1[0] = (int)((2u << 16) | (1u << 20) | (pi << 22) | (1u << 25));
                    // data_size=4B | pad_enable | pad_interval | pad_amount=2DW
      g1[1] = (int)((unsigned)K << 16);             // tensor_dim0[15:0]=K
      g1[2] = (int)(((unsigned)N & 0xffffu) << 16); // tensor_dim1[15:0]=N
      g1[3] = (int)(((unsigned)N >> 16) | ((unsigned)K << 16)); // dim1 hi | tile_dim0=K
      g1[4] = 16;                                   // tile_dim1=16, tile_dim2=0
      g1[5] = (int)(unsigned)K;                     // tensor_dim0_stride[31:0]=K
      g1[6] = 0;
      g1[7] = 0;
      v4i z4 = {0, 0, 0, 0};                        // dims >=3 unused
#if __has_include(<hip/amd_detail/amd_gfx1250_TDM.h>)
      v8i z8 = {0, 0, 0, 0, 0, 0, 0, 0};
      __builtin_amdgcn_tensor_load_to_lds(g0, g1, z4, z4, z8, 0);
#else
      __builtin_amdgcn_tensor_load_to_lds(g0, g1, z4, z4, 0);
#endif
      __builtin_amdgcn_s_wait_tensorcnt(0);
    }
    // TDM wrote sA behind the compiler's back: escaped-pointer clobber makes
    // the k-loop's fragment loads observe it (no undef-fold / CSE / hoist).
    asm volatile("" :: "v"(&sA[0]) : "memory");
#else
    for (int idx = threadIdx.x; idx < 16 * K; idx += 256) {  // exact multiple of 256
      int r = idx / K, c = idx - r * K;
      sA[r * lda + c] = X[(size_t)(row0 + r) * K + c];
    }
#endif
    __syncthreads();

    v8f acc = {};
    for (int kb = 0; kb < K; kb += 4) {
      const int kk = kb + 2 * half;
      v2f a, b;
      a.x = sA[mn * lda + kk + 0];
      a.y = sA[mn * lda + kk + 1];
      b.x = W[(size_t)(kk + 0) * HIDN + col0 + mn];
      b.y = W[(size_t)(kk + 1) * HIDN + col0 + mn];
      acc = __builtin_amdgcn_wmma_f32_16x16x4_f32(
          /*neg_a=*/false, a, /*neg_b=*/false, b,
          /*c_mod=*/(short)0, acc, /*reuse_a=*/false, /*reuse_b=*/false);
    }

    const float bv = bias ? bias[col0 + mn] : 0.f;
    for (int r = 0; r < 8; ++r)
      Y[(size_t)(row0 + r + 8 * half) * HIDN + col0 + mn] = acc[r] + bv;
    __syncthreads();   // sA readers done before next tile's TDM overwrites it
  }
}

// wa[k,h] = sum_d W[k, h*D+d] * av[h*D+d]   (folds attention vec through W)
__global__ void fold_attn(const float* __restrict__ W, const float* __restrict__ av,
                          float* __restrict__ wa, int H, int D)
{
  int idx = blockIdx.x * blockDim.x + threadIdx.x;
  if (idx >= HIDN * H) return;
  int k = idx / H, h = idx - k * H;
  float s = 0.f;
  for (int d = 0; d < D; ++d)
    s += W[(size_t)k * HIDN + h * D + d] * av[h * D + d];
  wa[k * H + h] = s;
}

// a[n,h] = X[n,:] @ wa[:,h]   one wave per node, 4 dims per lane
__global__ __launch_bounds__(256) void node_scores(
    const float* __restrict__ X, const float* __restrict__ wa,
    float* __restrict__ out, int N, int H)
{
  int node = blockIdx.x * 8 + (threadIdx.x >> 5);
  int lane = threadIdx.x & 31;
  if (node >= N) return;
  float4 x = *(const float4*)(X + (size_t)node * HIDN + lane * 4);
  for (int h = 0; h < H; ++h) {
    float p = x.x * wa[(lane * 4 + 0) * H + h] + x.y * wa[(lane * 4 + 1) * H + h]
            + x.z * wa[(lane * 4 + 2) * H + h] + x.w * wa[(lane * 4 + 3) * H + h];
    for (int o = 16; o > 0; o >>= 1) p += __shfl_xor(p, o, 32);
    if (lane == 0) out[(size_t)node * H + h] = p;
  }
}

__global__ void fill_i32(int* __restrict__ p, int v, int n) {
  int i = blockIdx.x * blockDim.x + threadIdx.x;
  if (i < n) p[i] = v;
}

// P1: e = lrelu(a_s[src]+a_d[dst]); segment max via int-encoded atomicMax
__global__ void edge_pass1(const int* __restrict__ src, const int* __restrict__ dst,
                           const float* __restrict__ as_, const float* __restrict__ ad_,
                           int* __restrict__ menc, int E, int H)
{
  int e = blockIdx.x * blockDim.x + threadIdx.x;
  if (e >= E) return;
  int s = src[e], d = dst[e];
  for (int h = 0; h < H; ++h) {
    float v = as_[s * H + h] + ad_[d * H + h];
    v = v > 0.f ? v : 0.2f * v;
    atomicMax(&menc[d * H + h], enc_f32(v));
  }
}

// P2: ex = exp(e - m[dst]); segment sum (recompute e, two cached gathers)
__global__ void edge_pass2(const int* __restrict__ src, const int* __restrict__ dst,
                           const float* __restrict__ as_, const float* __restrict__ ad_,
                           const int* __restrict__ menc, float* __restrict__ ssum,
                           int E, int H)
{
  int e = blockIdx.x * blockDim.x + threadIdx.x;
  if (e >= E) return;
  int s = src[e], d = dst[e];
  for (int h = 0; h < H; ++h) {
    float v = as_[s * H + h] + ad_[d * H + h];
    v = v > 0.f ? v : 0.2f * v;
    atom_add_f32(&ssum[d * H + h], expf(v - dec_f32(menc[d * H + h])));
  }
}

// P3: nOut[dst] += hs[src] * alpha.  One wave per edge, float4 per lane,
// 128 L2-resident f32 atomics per edge.
__global__ __launch_bounds__(256) void edge_pass3(
    const int* __restrict__ src, const int* __restrict__ dst,
    const float* __restrict__ hs, const float* __restrict__ as_,
    const float* __restrict__ ad_, const int* __restrict__ menc,
    const float* __restrict__ ssum, float* __restrict__ nOut,
    int E, int H, int D)
{
  int e = blockIdx.x * 8 + (threadIdx.x >> 5);
  if (e >= E) return;
  int lane = threadIdx.x & 31;
  int s = src[e], d = dst[e];
  int h = (lane * 4) / D;                    // head owning this lane's 4 dims
  float v = as_[s * H + h] + ad_[d * H + h];
  v = v > 0.f ? v : 0.2f * v;
  float alpha = expf(v - dec_f32(menc[d * H + h])) / (ssum[d * H + h] + 1e-16f);
  float4 x = *(const float4*)(hs + (size_t)s * HIDN + lane * 4);
  float* o = nOut + (size_t)d * HIDN + lane * 4;
  atom_add_f32(o + 0, x.x * alpha);
  atom_add_f32(o + 1, x.y * alpha);
  atom_add_f32(o + 2, x.z * alpha);
  atom_add_f32(o + 3, x.w * alpha);
}

// out = relu(LayerNorm(nbuf + bias)*g + b) + prev   (one wave per node)
__global__ __launch_bounds__(256) void epilogue_ln(
    const float* __restrict__ nbuf, const float* __restrict__ bias,
    const float* __restrict__ g, const float* __restrict__ b,
    const float* __restrict__ prev, float* __restrict__ out, int N)
{
  int node = blockIdx.x * 8 + (threadIdx.x >> 5);
  int lane = threadIdx.x & 31;
  if (node >= N) return;
  float4 v  = *(const float4*)(nbuf + (size_t)node * HIDN + lane * 4);
  float4 bb = *(const float4*)(bias + lane * 4);
  v.x += bb.x; v.y += bb.y; v.z += bb.z; v.w += bb.w;
  float sum = v.x + v.y + v.z + v.w;
  for (int o = 16; o > 0; o >>= 1) sum += __shfl_xor(sum, o, 32);
  float mu = sum * (1.f / 128.f);
  float dx = v.x - mu, dy = v.y - mu, dz = v.z - mu, dw = v.w - mu;
  float sq = dx * dx + dy * dy + dz * dz + dw * dw;
  for (int o = 16; o > 0; o >>= 1) sq += __shfl_xor(sq, o, 32);
  float inv = rsqrtf(sq * (1.f / 128.f) + 1e-5f);
  float4 gg = *(const float4*)(g + lane * 4);
  float4 be = *(const float4*)(b + lane * 4);
  float4 pv = *(const float4*)(prev + (size_t)node * HIDN + lane * 4);
  float4 r;
  r.x = fmaxf(dx * inv * gg.x + be.x, 0.f) + pv.x;
  r.y = fmaxf(dy * inv * gg.y + be.y, 0.f) + pv.y;
  r.z = fmaxf(dz * inv * gg.z + be.z, 0.f) + pv.z;
  r.w = fmaxf(dw * inv * gg.w + be.w, 0.f) + pv.w;
  *(float4*)(out + (size_t)node * HIDN + lane * 4) = r;
}

// ---------------------------------------------------------------------------
struct GatWs {
  float *hs, *a_s, *a_d, *ssum, *wa_s, *wa_d;
  int* menc;
};

static void run_gat(const float* xsrc, const float* xdst, const float* W,
                    const float* asv, const float* adv,
                    const int* src, const int* dst,
                    int H, int D, float* nOut, const GatWs& w, hipStream_t st)
{
  fold_attn<<<(HIDN * H + 255) / 256, 256, 0, st>>>(W, asv, w.wa_s, H, D);
  fold_attn<<<(HIDN * H + 255) / 256, 256, 0, st>>>(W, adv, w.wa_d, H, D);
  node_scores<<<(NNODE + 7) / 8, 256, 0, st>>>(xsrc, w.wa_s, w.a_s, NNODE, H);
  node_scores<<<(NNODE + 7) / 8, 256, 0, st>>>(xdst, w.wa_d, w.a_d, NNODE, H);
  gemm_wmma_f32<<<NNODE / 16, 256, 0, st>>>(xsrc, W, nullptr, w.hs, NNODE, HIDN);
  (void)hipMemsetAsync(w.ssum, 0, (size_t)NNODE * H * sizeof(float), st);
  fill_i32<<<(NNODE * H + 255) / 256, 256, 0, st>>>(w.menc, ENC_NEG_INF, NNODE * H);
  (void)hipMemsetAsync(nOut, 0, (size_t)NNODE * HIDN * sizeof(float), st);
  edge_pass1<<<(NEDGE + 255) / 256, 256, 0, st>>>(src, dst, w.a_s, w.a_d, w.menc, NEDGE, H);
  edge_pass2<<<(NEDGE + 255) / 256, 256, 0, st>>>(src, dst, w.a_s, w.a_d, w.menc, w.ssum, NEDGE, H);
  edge_pass3<<<(NEDGE + 7) / 8, 256, 0, st>>>(src, dst, w.hs, w.a_s, w.a_d,
                                              w.menc, w.ssum, nOut, NEDGE, H, D);
}

extern "C" void kernel_launch(void* const* d_in, const int* in_sizes, int n_in,
                              void* d_out, int out_size, void* d_ws, size_t ws_size,
                              hipStream_t stream)
{
  (void)in_sizes; (void)n_in; (void)out_size; (void)ws_size;
  const float* x_A  = (const float*)d_in[0];
  const float* x_B  = (const float*)d_in[1];
  const float* pWA  = (const float*)d_in[2];
  const float* pbA  = (const float*)d_in[3];
  const float* pWB  = (const float*)d_in[4];
  const float* pbB  = (const float*)d_in[5];
  const float* w0ab = (const float*)d_in[6];
  const float* as0ab= (const float*)d_in[7];
  const float* ad0ab= (const float*)d_in[8];
  const float* b0ab = (const float*)d_in[9];
  const float* w0ba = (const float*)d_in[10];
  const float* as0ba= (const float*)d_in[11];
  const float* ad0ba= (const float*)d_in[12];
  const float* b0ba = (const float*)d_in[13];
  const float* w1ab = (const float*)d_in[14];
  const float* as1ab= (const float*)d_in[15];
  const float* ad1ab= (const float*)d_in[16];
  const float* b1ab = (const float*)d_in[17];
  const float* w1ba = (const float*)d_in[18];
  const float* as1ba= (const float*)d_in[19];
  const float* ad1ba= (const float*)d_in[20];
  const float* b1ba = (const float*)d_in[21];
  const float* g0A  = (const float*)d_in[22];
  const float* bn0A = (const float*)d_in[23];
  const float* g0B  = (const float*)d_in[24];
  const float* bn0B = (const float*)d_in[25];
  const float* g1A  = (const float*)d_in[26];
  const float* bn1A = (const float*)d_in[27];
  const float* g1B  = (const float*)d_in[28];
  const float* bn1B = (const float*)d_in[29];
  const int* ei_AB  = (const int*)d_in[30];
  const int* ei_BA  = (const int*)d_in[31];
  const int* abSrc = ei_AB;           const int* abDst = ei_AB + NEDGE;
  const int* baSrc = ei_BA;           const int* baDst = ei_BA + NEDGE;

  const size_t NF = (size_t)NNODE * HIDN;
  float* outA = (float*)d_out;
  float* outB = outA + NF;

  // workspace layout (~132 MB)
  float* f = (float*)d_ws;
  float* hA = f; f += NF;
  float* hB = f; f += NF;
  float* hs = f; f += NF;
  float* nA = f; f += NF;
  float* nB = f; f += NF;
  GatWs w;
  w.hs  = hs;
  w.a_s = f; f += (size_t)NNODE * 4;
  w.a_d = f; f += (size_t)NNODE * 4;
  w.ssum= f; f += (size_t)NNODE * 4;
  w.menc= (int*)f; f += (size_t)NNODE * 4;
  w.wa_s= f; f += 512;
  w.wa_d= f; f += 512;

  // input projections
  gemm_wmma_f32<<<NNODE / 16, 256, 0, stream>>>(x_A, pWA, pbA, hA, NNODE, 128);
  gemm_wmma_f32<<<NNODE / 16, 256, 0, stream>>>(x_B, pWB, pbB, hB, NNODE, 64);

  // ---- layer 0 (H=4, D=32): both directions read hA/hB, write d_out halves
  run_gat(hA, hB, w0ab, as0ab, ad0ab, abSrc, abDst, 4, 32, nB, w, stream);
  run_gat(hB, hA, w0ba, as0ba, ad0ba, baSrc, baDst, 4, 32, nA, w, stream);
  epilogue_ln<<<NNODE / 8, 256, 0, stream>>>(nB, b0ab, g0B, bn0B, hB, outB, NNODE);
  epilogue_ln<<<NNODE / 8, 256, 0, stream>>>(nA, b0ba, g0A, bn0A, hA, outA, NNODE);

  // ---- layer 1 (H=1, D=128): reads d_out halves, final epilogue in place
  run_gat(outA, outB, w1ab, as1ab, ad1ab, abSrc, abDst, 1, 128, nB, w, stream);
  run_gat(outB, outA, w1ba, as1ba, ad1ba, baSrc, baDst, 1, 128, nA, w, stream);
  epilogue_ln<<<NNODE / 8, 256, 0, stream>>>(nB, b1ab, g1B, bn1B, outB, outB, NNODE);
  epilogue_ln<<<NNODE / 8, 256, 0, stream>>>(nA, b1ba, g1A, bn1A, outA, outA, NNODE);
}